// GaussianScene_90151363543777
// MI455X (gfx1250) — compile-verified
//
#include <hip/hip_runtime.h>

// Gaussian splatting, tiled rasterizer for MI455X (gfx1250, wave32).
// Grid: 16x16 tiles, block: 512 threads = 16 waves. Wave w owns tile column
// x=w (16 pixels). Color accumulation uses V_WMMA_F32_16X16X4_F32.
//
// B-operand colors live in a zero-padded pair-interleaved LDS array so the
// inner loop's B fetch is one unconditional ds_load_b64 per lane (no EXEC
// masking in the hot loop).

#define N_PTS   512
#define IMG_W   256
#define IMG_H   256
#define TILE_SZ 16
#define MIN_W   1e-6f

#define NSLOT   (N_PTS + 4)        // padded gaussian slots (multiple of 4)
#define NPAIR   (NSLOT / 2)        // gaussian pairs
#define COLB_SZ (NPAIR * 32)       // pair*32 + channel*2 + parity

typedef __attribute__((ext_vector_type(2))) float v2f;
typedef __attribute__((ext_vector_type(8))) float v8f;

__global__ __launch_bounds__(512)
void splat_tiles_kernel(const float* __restrict__ pts,    // [512,2]
                        const float* __restrict__ cols,   // [512,3]
                        const float* __restrict__ opac,   // [512]
                        const float* __restrict__ icov,   // [512,2,2]
                        const float* __restrict__ rad,    // [512]
                        float* __restrict__ out)          // [256,256,3]
{
    __shared__ float sGx[NSLOT], sGy[NSLOT];
    __shared__ float sA [NSLOT], sBC[NSLOT], sD[NSLOT], sOp[NSLOT];
    __shared__ float sColB[COLB_SZ];   // [pair][chan 0..15][parity], zero-padded
    __shared__ int   sWaveCnt[16], sWaveOff[16], sCount[2];

    const int tid   = threadIdx.x;
    const int lane  = tid & 31;
    const int wave  = tid >> 5;          // 0..15 : local pixel x
    const int ly    = lane & 15;         // lanes' row index / WMMA N
    const int half  = lane >> 4;         // 0: K=0,1   1: K=2,3
    const int tileX = blockIdx.x;
    const int tileY = blockIdx.y;
    const float tx0 = (float)(tileX * TILE_SZ);
    const float ty0 = (float)(tileY * TILE_SZ);

    // Reference "rendered" mask: last tile row/col are forced to zero.
    const bool rendered = (tx0 < (float)(IMG_W - TILE_SZ)) &&
                          (ty0 < (float)(IMG_H - TILE_SZ));
    if (!rendered) {
        if (tid < 256) {
            int lx = tid >> 4, lyy = tid & 15;
            int p = ((tileX * TILE_SZ + lx) * IMG_H + (tileY * TILE_SZ + lyy)) * 3;
            out[p + 0] = 0.f; out[p + 1] = 0.f; out[p + 2] = 0.f;
        }
        return;
    }

    // ---- Phase 0: zero the padded color array (channels 3..15 must be 0) --
    for (int i = tid; i < COLB_SZ; i += 512) sColB[i] = 0.f;

    // ---- Phase 1: cull + order-preserving compaction into LDS -------------
    const float gx = pts[tid * 2 + 0];
    const float gy = pts[tid * 2 + 1];
    const float rr = rad[tid];
    const bool ov = (floorf(gx - rr) <= tx0 + (float)TILE_SZ) &&
                    (ceilf (gx + rr) >= tx0) &&
                    (floorf(gy - rr) <= ty0 + (float)TILE_SZ) &&
                    (ceilf (gy + rr) >= ty0);

    const unsigned m32 = (unsigned)__ballot((int)ov);   // wave32: low 32 bits
    if (lane == 0) sWaveCnt[wave] = __popc(m32);
    __syncthreads();
    if (tid == 0) {
        int acc = 0;
        for (int i = 0; i < 16; ++i) { sWaveOff[i] = acc; acc += sWaveCnt[i]; }
        sCount[0] = acc;
        sCount[1] = (acc + 3) & ~3;                     // pad to multiple of 4
    }
    __syncthreads();
    const int n    = sCount[0];
    const int npad = sCount[1];

    if (ov) {
        int pos = sWaveOff[wave] + __popc(m32 & ((1u << lane) - 1u));
        sGx[pos] = gx;
        sGy[pos] = gy;
        sA [pos] = icov[tid * 4 + 0];
        sBC[pos] = icov[tid * 4 + 1] + icov[tid * 4 + 2];
        sD [pos] = icov[tid * 4 + 3];
        float o  = opac[tid];
        float s1 = 1.f / (1.f + __expf(-o));
        sOp[pos] = 1.f / (1.f + __expf(-s1));           // sigmoid(sigmoid(x))
        const int base = (pos >> 1) * 32 + (pos & 1);   // pair-interleaved
        sColB[base + 0] = cols[tid * 3 + 0];
        sColB[base + 2] = cols[tid * 3 + 1];
        sColB[base + 4] = cols[tid * 3 + 2];
    }
    if (tid >= n && tid < npad) {                       // alpha==0 padding
        sGx[tid] = 0.f; sGy[tid] = 0.f;
        sA[tid] = 0.f; sBC[tid] = 0.f; sD[tid] = 0.f; sOp[tid] = 0.f;
        // color padding already zero from Phase 0
    }
    __syncthreads();

    // ---- Phase 2: front-to-back compositing, 4 gaussians per WMMA ---------
    // Both half-waves run the same 16-pixel chain (duplicated), so the A
    // operand (16x4 f32: lanes 0-15 K0/K1, lanes 16-31 K2/K3) needs no
    // cross-lane traffic. T is monotone non-increasing, so the reference's
    // cumsum(trigger)==0 gate reduces to (T_after >= MIN_W).
    const float pxF = tx0 + (float)wave;
    const float pyF = ty0 + (float)ly;
    float T = 1.0f;
    v8f acc = {0.f, 0.f, 0.f, 0.f, 0.f, 0.f, 0.f, 0.f};

    for (int j = 0; j < npad; j += 4) {
        float w4[4];
#pragma unroll
        for (int kk = 0; kk < 4; ++kk) {
            const int g  = j + kk;
            const float dx = pxF - sGx[g];
            const float dy = pyF - sGy[g];
            const float q  = sA[g] * dx * dx + sBC[g] * dx * dy + sD[g] * dy * dy;
            const float alpha = sOp[g] * __expf(-0.5f * q);
            const float Tn = T * (1.f - alpha);
            w4[kk] = (Tn >= MIN_W) ? T * alpha : 0.f;
            T = Tn;
        }
        v2f a;
        a.x = half ? w4[2] : w4[0];
        a.y = half ? w4[3] : w4[1];
        // B fetch: one unconditional 8B LDS load -> (color[kb][c], color[kb+1][c])
        const int pair = (j >> 1) + half;               // kb = j + 2*half
        v2f b = *(const v2f*)&sColB[pair * 32 + ly * 2];
        // D(16x16 f32) += A(16x4) * B(4x16); cols 0..2 hold r,g,b sums.
        acc = __builtin_amdgcn_wmma_f32_16x16x4_f32(
                  false, a, false, b, (short)0, acc, false, false);

        if (__all(T < MIN_W)) break;                    // wave-uniform exit
    }

    // ---- Phase 3: writeback. D layout: VGPR r, lane l -> M=r+8*(l/16),
    // N=l%16. M = local pixel y, N = color channel.
    if (ly < 3) {
        const int xg = tileX * TILE_SZ + wave;
#pragma unroll
        for (int r8 = 0; r8 < 8; ++r8) {
            const int M = r8 + 8 * half;
            out[(xg * IMG_H + tileY * TILE_SZ + M) * 3 + ly] = acc[r8];
        }
    }
}

extern "C" void kernel_launch(void* const* d_in, const int* in_sizes, int n_in,
                              void* d_out, int out_size, void* d_ws, size_t ws_size,
                              hipStream_t stream) {
    (void)in_sizes; (void)n_in; (void)d_ws; (void)ws_size; (void)out_size;
    const float* pts  = (const float*)d_in[0];
    const float* cols = (const float*)d_in[1];
    const float* opac = (const float*)d_in[2];
    const float* icov = (const float*)d_in[3];
    const float* rad  = (const float*)d_in[4];
    // d_in[5..7] = width/height/tile_size scalars (compile-time constants here)
    dim3 grid(IMG_W / TILE_SZ, IMG_H / TILE_SZ);
    splat_tiles_kernel<<<grid, 512, 0, stream>>>(pts, cols, opac, icov, rad,
                                                 (float*)d_out);
}